// ShiftingLayer_15487652069664
// MI455X (gfx1250) — compile-verified
//
#include <hip/hip_runtime.h>

#define SL_H 4096
#define SL_W 8192

typedef float v4f __attribute__((ext_vector_type(4)));

// ---------------------------------------------------------------------------
// Kernel 1: zero-fill the output with regular-temporal (RT) B128 stores so the
// 128 MB output allocates in the 192 MB L2 and stays resident for the scatter.
// ---------------------------------------------------------------------------
__global__ void sl_zero_out(v4f* __restrict__ out) {
    size_t idx = (size_t)blockIdx.x * blockDim.x + threadIdx.x;
    out[idx] = (v4f){0.0f, 0.0f, 0.0f, 0.0f};
}

// ---------------------------------------------------------------------------
// Kernel 2: scatter pass.
//   - 2D grid: blockIdx.y = row (no integer division for i/j)
//   - each thread owns 4 consecutive columns -> 3x global_load_b128 th:NT
//     (inputs are streamed exactly once; NT keeps them from evicting the
//      L2-resident output lines)
//   - (int) cast of fp32 truncates toward zero (v_cvt_i32_f32), matching
//     torch .int() / jnp.trunc
//   - unsigned compare implements mode="drop" incl. negative indices
//   - output stores are RT so they hit the L2-resident zero-filled lines
// ---------------------------------------------------------------------------
__global__ void sl_scatter_shift(const float* __restrict__ x,
                                 const float* __restrict__ wr,
                                 const float* __restrict__ wc,
                                 float* __restrict__ out) {
    const int i  = blockIdx.y;
    const int j0 = (blockIdx.x * blockDim.x + threadIdx.x) * 4;
    const size_t base = (size_t)i * SL_W + (size_t)j0;

    const v4f xv = __builtin_nontemporal_load((const v4f*)(x  + base));
    const v4f rv = __builtin_nontemporal_load((const v4f*)(wr + base));
    const v4f cv = __builtin_nontemporal_load((const v4f*)(wc + base));

#pragma unroll
    for (int k = 0; k < 4; ++k) {
        const int r = i        + (int)rv[k];   // trunc toward zero
        const int c = (j0 + k) + (int)cv[k];   // trunc toward zero
        if ((unsigned)r < (unsigned)SL_H && (unsigned)c < (unsigned)SL_W) {
            out[(size_t)r * SL_W + (unsigned)c] = xv[k];
        }
    }
}

extern "C" void kernel_launch(void* const* d_in, const int* in_sizes, int n_in,
                              void* d_out, int out_size, void* d_ws, size_t ws_size,
                              hipStream_t stream) {
    (void)in_sizes; (void)n_in; (void)out_size; (void)d_ws; (void)ws_size;

    const float* x  = (const float*)d_in[0];
    const float* wr = (const float*)d_in[1];
    const float* wc = (const float*)d_in[2];
    float*      out = (float*)d_out;

    // Zero-fill: 4096*8192 floats = 8,388,608 float4s; 256 threads/block.
    {
        const int threads = 256;
        const int nvec4   = (SL_H * SL_W) / 4;       // 8,388,608
        const int blocks  = nvec4 / threads;         // 32,768
        sl_zero_out<<<blocks, threads, 0, stream>>>((v4f*)out);
    }

    // Scatter: each thread handles 4 columns of one row.
    {
        const int threads = 256;                     // 8 wave32 per block
        dim3 grid(SL_W / (threads * 4), SL_H, 1);    // (8, 4096)
        sl_scatter_shift<<<grid, dim3(threads, 1, 1), 0, stream>>>(x, wr, wc, out);
    }
}